// SkipGram_35330400977147
// MI455X (gfx1250) — compile-verified
//
#include <hip/hip_runtime.h>
#include <hip/hip_bf16.h>

typedef __attribute__((ext_vector_type(2))) float v2f;
typedef __attribute__((ext_vector_type(8))) float v8f;

#define D_DIM 128
#define K_NEG 20
#define WAVES_PER_BLOCK 4
#define BLOCKS 256
#define BLOCK_THREADS (WAVES_PER_BLOCK * 32)

__device__ __forceinline__ float loss_term(float x) {
    // -log(sigmoid(x) + 1e-8); fast v_rcp_f32 instead of IEEE divide chain
    // (eps=1e-8 in the reference dwarfs the ~1 ulp rcp error).
    float e = __expf(-x);
    float s = __builtin_amdgcn_rcpf(1.0f + e);
    return -__logf(s + 1e-8f);
}

__device__ __forceinline__ float read_lane_f(float v, int lane) {
    return __int_as_float(__builtin_amdgcn_readlane(__float_as_int(v), lane));
}

// Accumulate Gram-diagonal loss terms from a 16x16 f32 WMMA accumulator.
// C/D layout: VGPR r, lanes 0-15 -> (M=r, N=lane); lanes 16-31 -> (M=8+r, N=lane-16).
// Diagonal (i,i): i<8 at (reg=i, lane=i); i>=8 at (reg=i-8, lane=i+16).
__device__ __forceinline__ float diag_loss(v8f acc, bool negate) {
    float s = 0.0f;
#pragma unroll
    for (int i = 0; i < 16; ++i) {
        int reg  = i & 7;
        int lane = (i < 8) ? i : (i + 16);
        float v = read_lane_f(acc[reg], lane);
        s += loss_term(negate ? -v : v);
    }
    return s;
}

__global__ void __launch_bounds__(BLOCK_THREADS)
skipgram_loss_kernel(const int* __restrict__ center,
                     const int* __restrict__ context,
                     const int* __restrict__ negatives,
                     const float* __restrict__ center_emb,
                     const float* __restrict__ context_emb,
                     float* __restrict__ partials,
                     int B) {
    __shared__ float lds_vc[WAVES_PER_BLOCK * 16 * D_DIM]; // 32 KB: 16 vc rows per wave
    __shared__ float wave_part[WAVES_PER_BLOCK];

    const int lane       = threadIdx.x & 31;
    const int wave_in_bl = threadIdx.x >> 5;
    const int wave_glob  = blockIdx.x * WAVES_PER_BLOCK + wave_in_bl;
    const int n_waves    = gridDim.x * WAVES_PER_BLOCK;
    const int laneN      = lane & 15;          // M index (A) and N index (B) for this lane
    const int kOff       = (lane >> 4) << 1;   // K offset within 4-wide chunk: 0 or 2

    float* vcs = &lds_vc[wave_in_bl * 16 * D_DIM];

    float loss = 0.0f;
    const int n_groups = B >> 4;               // 16 batch elements per group

    for (int g = wave_glob; g < n_groups; g += n_waves) {
        const int bN  = g * 16 + laneN;
        const int ctr = center[bN];
        const int ctx = context[bN];

        __syncthreads(); // protect LDS reuse across group iterations (uniform trip count)

        // Stage the 16 vc rows of this group into LDS (coalesced 512B per row).
#pragma unroll 4
        for (int r = 0; r < 16; ++r) {
            int row = __builtin_amdgcn_readlane(ctr, r);
            const float4* src = (const float4*)(center_emb + (size_t)row * D_DIM + lane * 4);
            *(float4*)(vcs + r * D_DIM + lane * 4) = *src;
        }
        __syncthreads();

        const float* aRow = vcs + laneN * D_DIM; // A fragment source: vc[b_laneN] (LDS)

        // ---- positive term: diag( Vc * Vp^T ) over D in chunks of 4 ----
        {
            const float* bRow = context_emb + (size_t)ctx * D_DIM;
            v8f acc = {};
#pragma unroll 4
            for (int c = 0; c < 32; ++c) {
                v2f a = *(const v2f*)(aRow + 4 * c + kOff);
                v2f b = *(const v2f*)(bRow + 4 * c + kOff);
                acc = __builtin_amdgcn_wmma_f32_16x16x4_f32(
                    false, a, false, b, (short)0, acc, false, false);
            }
            loss += diag_loss(acc, /*negate=*/false);
        }

        // ---- negative terms: diag( Vc * Vn_k^T ) for k = 0..K-1 ----
        for (int k = 0; k < K_NEG; ++k) {
            const int ni = negatives[(size_t)bN * K_NEG + k];
            const float* bRow = context_emb + (size_t)ni * D_DIM;
            v8f acc = {};
#pragma unroll 4
            for (int c = 0; c < 32; ++c) {
                v2f a = *(const v2f*)(aRow + 4 * c + kOff);
                v2f b = *(const v2f*)(bRow + 4 * c + kOff);
                acc = __builtin_amdgcn_wmma_f32_16x16x4_f32(
                    false, a, false, b, (short)0, acc, false, false);
            }
            loss += diag_loss(acc, /*negate=*/true);
        }
    }

    // loss is wave-uniform (built from readlane broadcasts); deposit one value per wave.
    __syncthreads();
    if (lane == 0) wave_part[wave_in_bl] = loss;
    __syncthreads();
    if (threadIdx.x == 0) {
        float s = 0.0f;
#pragma unroll
        for (int w = 0; w < WAVES_PER_BLOCK; ++w) s += wave_part[w];
        partials[blockIdx.x] = s;
    }
}

__global__ void __launch_bounds__(256)
final_reduce_kernel(const float* __restrict__ partials, float* __restrict__ out,
                    int n, float inv_b) {
    __shared__ float red[256];
    const int t = threadIdx.x;
    red[t] = (t < n) ? partials[t] : 0.0f;
    __syncthreads();
#pragma unroll
    for (int s = 128; s > 0; s >>= 1) {
        if (t < s) red[t] += red[t + s];
        __syncthreads();
    }
    if (t == 0) out[0] = red[0] * inv_b;
}

extern "C" void kernel_launch(void* const* d_in, const int* in_sizes, int n_in,
                              void* d_out, int out_size, void* d_ws, size_t ws_size,
                              hipStream_t stream) {
    const int*   center      = (const int*)d_in[0];
    const int*   context     = (const int*)d_in[1];
    const int*   negatives   = (const int*)d_in[2];
    const float* center_emb  = (const float*)d_in[3];
    const float* context_emb = (const float*)d_in[4];
    float*       out         = (float*)d_out;
    float*       partials    = (float*)d_ws;   // BLOCKS floats

    const int B = in_sizes[0];

    skipgram_loss_kernel<<<BLOCKS, BLOCK_THREADS, 0, stream>>>(
        center, context, negatives, center_emb, context_emb, partials, B);

    final_reduce_kernel<<<1, 256, 0, stream>>>(partials, out, BLOCKS, 1.0f / (float)B);
}